// Maximum_Mean_Discrepancy_71270687310347
// MI455X (gfx1250) — compile-verified
//
#include <hip/hip_runtime.h>
#include <math.h>

typedef __attribute__((ext_vector_type(2))) float v2f;
typedef __attribute__((ext_vector_type(8))) float v8f;

#define N_ROWS 8192
#define B_ROWS 4096
#define KDIM   512

// workspace layout (float indices)
#define WS_SQ    0        // 8192 row squared-norms
#define WS_CS    8192     // 512 column sums
#define WS_SCALE 8704     // 1 scalar: 1/(16*bandwidth)
#define WS_PART  8705     // 4096 per-block partial sums
#define N_PART   4096     // 64x64 block grid

// ---------------- stats: row squared norms ----------------
__global__ void row_sq_kernel(const float* __restrict__ src,
                              const float* __restrict__ tgt,
                              float* __restrict__ ws) {
    int gid = blockIdx.x * blockDim.x + threadIdx.x;
    if (gid >= N_ROWS) return;
    const float* row = (gid < B_ROWS) ? (src + (size_t)gid * KDIM)
                                      : (tgt + (size_t)(gid - B_ROWS) * KDIM);
    float acc = 0.f;
    for (int k = 0; k < KDIM; k += 4) {
        float4 v = *(const float4*)(row + k);
        acc += v.x * v.x + v.y * v.y + v.z * v.z + v.w * v.w;
    }
    ws[WS_SQ + gid] = acc;
}

// ---------------- stats: column sums (coalesced across threads) ----------------
__global__ void col_sum_kernel(const float* __restrict__ src,
                               const float* __restrict__ tgt,
                               float* __restrict__ ws) {
    int k = blockIdx.x * blockDim.x + threadIdx.x;
    if (k >= KDIM) return;
    float acc = 0.f;
    for (int i = 0; i < B_ROWS; ++i) acc += src[(size_t)i * KDIM + k];
    for (int i = 0; i < B_ROWS; ++i) acc += tgt[(size_t)i * KDIM + k];
    ws[WS_CS + k] = acc;
}

// ---------------- bandwidth from closed form:
// sum(l2) = 2*n*sum(sq) - 2*||colsum||^2 ; bw = sum(l2)/(n^2-n)/4 ----------------
__global__ void bandwidth_kernel(float* __restrict__ ws) {
    __shared__ float r1[256];
    __shared__ float r2[256];
    int tid = threadIdx.x;
    float s1 = 0.f, s2 = 0.f;
    for (int i = tid; i < N_ROWS; i += 256) s1 += ws[WS_SQ + i];
    for (int i = tid; i < KDIM; i += 256) { float c = ws[WS_CS + i]; s2 += c * c; }
    r1[tid] = s1; r2[tid] = s2;
    __syncthreads();
    for (int off = 128; off > 0; off >>= 1) {
        if (tid < off) { r1[tid] += r1[tid + off]; r2[tid] += r2[tid + off]; }
        __syncthreads();
    }
    if (tid == 0) {
        float n = (float)N_ROWS;
        float suml2 = 2.f * n * r1[0] - 2.f * r2[0];
        float bw = suml2 / (n * n - n);
        bw *= 0.25f;                       // / KERNEL_MUL^(KERNEL_NUM//2) = /4
        ws[WS_SCALE] = 1.f / (16.f * bw);  // smallest-exponent scale
    }
}

// ---------------- fused gram-tile + kernel-sum ----------------
// Each wave32 computes one 64x64 tile of total@total^T via 4x4 grid of
// V_WMMA_F32_16X16X4_F32, then fuses l2 -> 5 exponentials -> signed sum.
// Only lower-triangle tiles computed; off-diagonal weighted x2.
__global__ __launch_bounds__(128)
void mmd_tile_kernel(const float* __restrict__ src,
                     const float* __restrict__ tgt,
                     float* __restrict__ ws) {
    __shared__ float wsum[4];
    const int lane   = threadIdx.x & 31;
    const int wave   = threadIdx.x >> 5;
    const int laneLo = lane & 15;
    const int khalf  = lane >> 4;    // 0: K 0,1   1: K 2,3
    const int koff   = khalf * 2;

    // 128x128 block tile, 64x64 per wave
    const int wm = blockIdx.y * 128 + (wave >> 1) * 64;
    const int wn = blockIdx.x * 128 + (wave & 1) * 64;

    float partial = 0.f;
    if (wn <= wm) {   // lower triangle only (tiles never straddle the 4096 split)
        const float sign   = ((wm < B_ROWS) == (wn < B_ROWS)) ? 1.f : -1.f;
        const float weight = (wn < wm) ? 2.f : 1.f;

        const float* Abase = (wm < B_ROWS) ? (src + (size_t)wm * KDIM)
                                           : (tgt + (size_t)(wm - B_ROWS) * KDIM);
        const float* Bbase = (wn < B_ROWS) ? (src + (size_t)wn * KDIM)
                                           : (tgt + (size_t)(wn - B_ROWS) * KDIM);

        const float* Arow[4];
        const float* Brow[4];
#pragma unroll
        for (int i = 0; i < 4; ++i) {
            Arow[i] = Abase + (size_t)(i * 16 + laneLo) * KDIM + koff;
            Brow[i] = Bbase + (size_t)(i * 16 + laneLo) * KDIM + koff;
        }

        v8f acc[4][4];
        {
            v8f z = {};
#pragma unroll
            for (int mi = 0; mi < 4; ++mi)
#pragma unroll
                for (int ni = 0; ni < 4; ++ni) acc[mi][ni] = z;
        }

#pragma unroll 2
        for (int k0 = 0; k0 < KDIM; k0 += 4) {
            v2f a[4], b[4];
#pragma unroll
            for (int i = 0; i < 4; ++i) a[i] = *(const v2f*)(Arow[i] + k0);
#pragma unroll
            for (int i = 0; i < 4; ++i) b[i] = *(const v2f*)(Brow[i] + k0);
#pragma unroll
            for (int mi = 0; mi < 4; ++mi)
#pragma unroll
                for (int ni = 0; ni < 4; ++ni)
                    acc[mi][ni] = __builtin_amdgcn_wmma_f32_16x16x4_f32(
                        false, a[mi], false, b[ni],
                        (short)0, acc[mi][ni], false, false);
        }

        // epilogue: l2 -> t=exp(-l2/(16 bw)) -> t + t^2 + t^4 + t^8 + t^16
        const float inv16bw = ws[WS_SCALE];
        float sqr[4][8];
#pragma unroll
        for (int mi = 0; mi < 4; ++mi) {
            int base = wm + mi * 16 + khalf * 8;  // C layout: VGPR v -> M = v (+8 hi half)
#pragma unroll
            for (int v = 0; v < 8; ++v) sqr[mi][v] = ws[WS_SQ + base + v];
        }
#pragma unroll
        for (int ni = 0; ni < 4; ++ni) {
            float sqc = ws[WS_SQ + wn + ni * 16 + laneLo];  // C layout: N = lane&15
#pragma unroll
            for (int mi = 0; mi < 4; ++mi) {
#pragma unroll
                for (int v = 0; v < 8; ++v) {
                    float l2 = fmaxf(sqr[mi][v] + sqc - 2.f * acc[mi][ni][v], 0.f);
                    float t  = __expf(-l2 * inv16bw);
                    float t2 = t * t, t4 = t2 * t2, t8 = t4 * t4, t16 = t8 * t8;
                    partial += t + t2 + t4 + t8 + t16;
                }
            }
        }
        partial *= sign * weight;
    }

    // wave reduce (wave32), then block reduce -> deterministic per-block partial
    for (int off = 16; off > 0; off >>= 1)
        partial += __shfl_xor(partial, off, 32);
    if (lane == 0) wsum[wave] = partial;
    __syncthreads();
    if (threadIdx.x == 0)
        ws[WS_PART + blockIdx.y * gridDim.x + blockIdx.x] =
            wsum[0] + wsum[1] + wsum[2] + wsum[3];
}

// ---------------- fixed-order final reduction ----------------
__global__ void final_reduce_kernel(const float* __restrict__ ws,
                                    float* __restrict__ out) {
    __shared__ float r[256];
    int tid = threadIdx.x;
    float a = 0.f;
    for (int i = tid; i < N_PART; i += 256) a += ws[WS_PART + i];
    r[tid] = a;
    __syncthreads();
    for (int off = 128; off > 0; off >>= 1) {
        if (tid < off) r[tid] += r[tid + off];
        __syncthreads();
    }
    if (tid == 0) out[0] = r[0] * (1.f / ((float)B_ROWS * (float)B_ROWS));
}

extern "C" void kernel_launch(void* const* d_in, const int* in_sizes, int n_in,
                              void* d_out, int out_size, void* d_ws, size_t ws_size,
                              hipStream_t stream) {
    const float* src = (const float*)d_in[0];
    const float* tgt = (const float*)d_in[1];
    float* ws  = (float*)d_ws;
    float* out = (float*)d_out;

    row_sq_kernel<<<N_ROWS / 256, 256, 0, stream>>>(src, tgt, ws);
    col_sum_kernel<<<KDIM / 256, 256, 0, stream>>>(src, tgt, ws);
    bandwidth_kernel<<<1, 256, 0, stream>>>(ws);

    dim3 grid(64, 64);   // 128x128 output region per block
    mmd_tile_kernel<<<grid, 128, 0, stream>>>(src, tgt, ws);

    final_reduce_kernel<<<1, 256, 0, stream>>>(ws, out);
}